// PolicyEncoder_79044578116211
// MI455X (gfx1250) — compile-verified
//
#include <hip/hip_runtime.h>
#include <hip/hip_bf16.h>

// PolicyEncoder: out[n,:] = bias + w_state0[obs0[n]] + w_state1[obs1[n]]
//                         + w_act0[act0[n]] + w_act1[act1[n]]
// N = 262144 rows, D = 128 floats (512 B per row).
//
// Memory-bound gather+add. MI455X model:
//   - tables total ~78 MB -> resident in 192 MB L2 after warm-up
//   - output 128 MB streamed once -> non-temporal stores (don't evict tables)
//   - index arrays 4 MB streamed once -> non-temporal loads
//   - HBM floor ~ 128 MB / 23.3 TB/s ~ 5.5 us (+ cold table misses)
// One wave32 per row: lane l handles columns [4l, 4l+4) = one b128 per table.
// Indices are software-pipelined one iteration ahead: gathers issue with no
// load->address dependency at loop top, and the next rows of the two large
// tables are prefetched (global_prefetch_b8) a full iteration early.

typedef float v4f __attribute__((ext_vector_type(4)));

#define POLENC_D 128

__global__ __launch_bounds__(256) void PolicyEncoder_gather_kernel(
    const int* __restrict__ obs0, const int* __restrict__ obs1,
    const int* __restrict__ act0, const int* __restrict__ act1,
    const float* __restrict__ w_state0, const float* __restrict__ w_state1,
    const float* __restrict__ w_act0,  const float* __restrict__ w_act1,
    const float* __restrict__ bias, float* __restrict__ out, int nrows)
{
    const int lane          = threadIdx.x & 31;          // wave32 lane
    const int widInBlock    = threadIdx.x >> 5;
    const int wavesPerBlock = blockDim.x >> 5;
    const int wave          = blockIdx.x * wavesPerBlock + widInBlock;
    const int waveStride    = gridDim.x * wavesPerBlock;
    const int col           = lane << 2;                 // 4 floats per lane

    // Bias chunk for this lane (512 B total, always cache-hot). Hoisted.
    const v4f bvec = *(const v4f*)(bias + col);

    int row = wave;
    if (row >= nrows) return;

    // Prologue: indices for the first row. Wave-uniform -> SGPRs.
    // NT loads: index arrays are streamed once; keep L2 for the tables.
    int i0 = __builtin_amdgcn_readfirstlane(__builtin_nontemporal_load(obs0 + row));
    int i1 = __builtin_amdgcn_readfirstlane(__builtin_nontemporal_load(obs1 + row));
    int i2 = __builtin_amdgcn_readfirstlane(__builtin_nontemporal_load(act0 + row));
    int i3 = __builtin_amdgcn_readfirstlane(__builtin_nontemporal_load(act1 + row));

    for (;;) {
        const int  next     = row + waveStride;
        const bool has_next = next < nrows;
        int n0 = 0, n1 = 0, n2 = 0, n3 = 0;
        if (has_next) {
            // Load next iteration's indices now (used as this iteration's
            // prefetch targets and carried in SGPRs to the next iteration).
            n0 = __builtin_amdgcn_readfirstlane(__builtin_nontemporal_load(obs0 + next));
            n1 = __builtin_amdgcn_readfirstlane(__builtin_nontemporal_load(obs1 + next));
            n2 = __builtin_amdgcn_readfirstlane(__builtin_nontemporal_load(act0 + next));
            n3 = __builtin_amdgcn_readfirstlane(__builtin_nontemporal_load(act1 + next));
            // Prefetch next rows of the two LARGE tables only
            // (w_state0: 51 MB, w_state1: 26 MB; the 512 KB act tables are
            // always L2-resident).
            __builtin_prefetch(w_state0 + (size_t)n0 * POLENC_D + col, 0, 3);
            __builtin_prefetch(w_state1 + (size_t)n1 * POLENC_D + col, 0, 3);
        }

        // Four coalesced 512 B row gathers (global_load_b128 per lane);
        // addresses were ready at loop entry -> no index-load stall here.
        const v4f a0 = *(const v4f*)(w_state0 + (size_t)i0 * POLENC_D + col);
        const v4f a1 = *(const v4f*)(w_state1 + (size_t)i1 * POLENC_D + col);
        const v4f a2 = *(const v4f*)(w_act0  + (size_t)i2 * POLENC_D + col);
        const v4f a3 = *(const v4f*)(w_act1  + (size_t)i3 * POLENC_D + col);

        v4f acc = bvec + a0;
        acc = acc + a1;
        acc = acc + a2;
        acc = acc + a3;

        // Non-temporal 128-bit store: output is written once, never re-read.
        __builtin_nontemporal_store(acc, (v4f*)(out + (size_t)row * POLENC_D + col));

        if (!has_next) break;
        row = next;
        i0 = n0; i1 = n1; i2 = n2; i3 = n3;
    }
}

extern "C" void kernel_launch(void* const* d_in, const int* in_sizes, int n_in,
                              void* d_out, int out_size, void* d_ws, size_t ws_size,
                              hipStream_t stream) {
    // setup_inputs() order:
    // 0 obs0[N] i32, 1 obs1[N] i32, 2 act0[N] i32, 3 act1[N] i32,
    // 4 w_state0[100000,128] f32, 5 w_state1[50000,128] f32,
    // 6 w_act0[1000,128] f32, 7 w_act1[1000,128] f32, 8 bias[128] f32
    const int*   obs0     = (const int*)d_in[0];
    const int*   obs1     = (const int*)d_in[1];
    const int*   act0     = (const int*)d_in[2];
    const int*   act1     = (const int*)d_in[3];
    const float* w_state0 = (const float*)d_in[4];
    const float* w_state1 = (const float*)d_in[5];
    const float* w_act0   = (const float*)d_in[6];
    const float* w_act1   = (const float*)d_in[7];
    const float* bias     = (const float*)d_in[8];
    float*       out      = (float*)d_out;

    const int nrows = in_sizes[0];            // N = 262144

    constexpr int BLOCK = 256;                // 8 wave32 per block
    constexpr int ROWS_PER_WAVE = 8;          // grid-stride depth (pipelining)
    const int wavesPerBlock = BLOCK / 32;
    const int wavesNeeded   = (nrows + ROWS_PER_WAVE - 1) / ROWS_PER_WAVE;
    int blocks = (wavesNeeded + wavesPerBlock - 1) / wavesPerBlock;
    if (blocks < 1) blocks = 1;

    PolicyEncoder_gather_kernel<<<blocks, BLOCK, 0, stream>>>(
        obs0, obs1, act0, act1,
        w_state0, w_state1, w_act0, w_act1,
        bias, out, nrows);
}